// DGLAIAConv_46943992545633
// MI455X (gfx1250) — compile-verified
//
#include <hip/hip_runtime.h>
#include <hip/hip_bf16.h>
#include <math.h>

// ---------------------------------------------------------------------------
// DGL-AIA conv for MI455X (gfx1250): fp32 WMMA projections + L2-resident
// gather/scatter edge passes. Problem is L2-bandwidth/atomic bound, so fp32
// is the right precision; WMMA 16x16x4 F32 keeps exact accuracy while using
// the matrix pipes for the 4 node projections.
// ---------------------------------------------------------------------------

typedef __attribute__((ext_vector_type(2))) float v2f;
typedef __attribute__((ext_vector_type(8))) float v8f;

// ---- helpers ---------------------------------------------------------------

__device__ __forceinline__ unsigned encF(float f) {
    unsigned u = __float_as_uint(f);
    return (u & 0x80000000u) ? ~u : (u | 0x80000000u);   // order-preserving
}
__device__ __forceinline__ float decF(unsigned e) {
    return __uint_as_float((e & 0x80000000u) ? (e & 0x7FFFFFFFu) : ~e);
}
__device__ __forceinline__ void atomicMaxU(unsigned* p, unsigned v) {
    __hip_atomic_fetch_max(p, v, __ATOMIC_RELAXED, __HIP_MEMORY_SCOPE_AGENT);
}
__device__ __forceinline__ void atomicAddF(float* p, float v) {
    __hip_atomic_fetch_add(p, v, __ATOMIC_RELAXED, __HIP_MEMORY_SCOPE_AGENT);
}

// ---- init: out = x_dst, zero accumulators ---------------------------------

__global__ void init_kernel(const float* __restrict__ x_dst, float* __restrict__ out,
                            float* __restrict__ g, unsigned* __restrict__ nodeMax,
                            float* __restrict__ nodeSum, int nd, int nh) {
    int i = blockIdx.x * blockDim.x + threadIdx.x;
    if (i < nd) { out[i] = x_dst[i]; g[i] = 0.f; }
    if (i < nh) { nodeMax[i] = 0u; nodeSum[i] = 0.f; }
}

// ---- projections via V_WMMA_F32_16X16X4_F32 -------------------------------
// One wave computes one 16x16 tile of Out = X(16x64) * W^T(64x16).
// A (16x4 f32): lanes 0-15 hold rows M, kh=lane>>4 selects K pair {0,1}/{2,3}.
// B (4x16 f32): mirrored layout, lane&15 = column, kh selects K pair.
// C (16x16 f32): 8 VGPRs; lanes 0-15 -> M=vgpr, lanes 16-31 -> M=vgpr+8.

__global__ __launch_bounds__(32)
void proj_wmma_kernel(const float* __restrict__ x_src, const float* __restrict__ x_dst,
                      const float* __restrict__ Wq, const float* __restrict__ Wk,
                      const float* __restrict__ Wv, const float* __restrict__ Wt,
                      float* __restrict__ Qo, float* __restrict__ Ko,
                      float* __restrict__ Vo, float* __restrict__ To, int Nn) {
    const int lane    = threadIdx.x;
    const int rowTile = blockIdx.x;
    const int colTile = blockIdx.y & 3;
    const int mat     = blockIdx.y >> 2;

    const float* X; const float* W; float* O;
    switch (mat) {
        case 0:  X = x_dst; W = Wq; O = Qo; break;
        case 1:  X = x_src; W = Wk; O = Ko; break;
        case 2:  X = x_src; W = Wv; O = Vo; break;
        default: X = x_src; W = Wt; O = To; break;
    }

    const int r   = lane & 15;
    const int kh  = lane >> 4;                 // 0/1: which K sub-pair
    int row = rowTile * 16 + r;
    if (row >= Nn) row = Nn - 1;               // clamp loads; EXEC stays all-ones
    const int col = colTile * 16 + r;

    v8f acc = {0.f, 0.f, 0.f, 0.f, 0.f, 0.f, 0.f, 0.f};
    #pragma unroll
    for (int kk = 0; kk < 64; kk += 4) {
        const int k0 = kk + 2 * kh;
        v2f a, b;
        a.x = X[row * 64 + k0];  a.y = X[row * 64 + k0 + 1];
        b.x = W[col * 64 + k0];  b.y = W[col * 64 + k0 + 1];   // W^T[k, col]
        acc = __builtin_amdgcn_wmma_f32_16x16x4_f32(
                  false, a, false, b, (short)0, acc, false, false);
    }

    const int orow = rowTile * 16 + kh * 8;
    #pragma unroll
    for (int i = 0; i < 8; ++i) {
        int rr = orow + i;
        if (rr < Nn) O[rr * 64 + col] = acc[i];
    }
}

// ---- qe[n][h] = sum_j We[h*16+j] * Q[n][h*16+j] ---------------------------

__global__ void qe_kernel(const float* __restrict__ Q, const float* __restrict__ We,
                          float* __restrict__ qe, int total) {
    int i = blockIdx.x * blockDim.x + threadIdx.x;
    if (i >= total) return;
    const int h = i & 3;
    const float* q = Q + i * 16;       // (n*4+h)*16 == n*64 + h*16
    const float* w = We + h * 16;
    float s = 0.f;
    #pragma unroll
    for (int j = 0; j < 16; ++j) s += q[j] * w[j];
    qe[i] = s;
}

// ---- pass A: logits + segment max -----------------------------------------

__global__ void edge_logits_kernel(const float* __restrict__ K, const float* __restrict__ Q,
                                   const float* __restrict__ qe, const float* __restrict__ ef,
                                   const int* __restrict__ src, const int* __restrict__ dst,
                                   float* __restrict__ eaBuf, unsigned* __restrict__ nodeMax,
                                   int total) {
    int i = blockIdx.x * blockDim.x + threadIdx.x;
    if (i >= total) return;
    const int e = i >> 2, h = i & 3;
    const int s = src[e], dn = dst[e];
    const float4* kp = (const float4*)(K + s * 64 + h * 16);
    const float4* qp = (const float4*)(Q + dn * 64 + h * 16);
    float dot = 0.f;
    #pragma unroll
    for (int j = 0; j < 4; ++j) {
        float4 kv = kp[j], qv = qp[j];
        dot += kv.x * qv.x + kv.y * qv.y + kv.z * qv.z + kv.w * qv.w;
    }
    float ev = (dot + ef[e] * qe[dn * 4 + h]) * 0.25f;   // 1/sqrt(d), d=16
    eaBuf[i] = ev;
    atomicMaxU(nodeMax + dn * 4 + h, encF(ev));
}

// ---- pass B: exp(e - max) + segment sum -----------------------------------

__global__ void edge_exp_kernel(const int* __restrict__ dst, const unsigned* __restrict__ nodeMax,
                                float* __restrict__ eaBuf, float* __restrict__ nodeSum, int total) {
    int i = blockIdx.x * blockDim.x + threadIdx.x;
    if (i >= total) return;
    const int e = i >> 2, h = i & 3;
    const int dn = dst[e];
    float ex = __expf(eaBuf[i] - decF(nodeMax[dn * 4 + h]));
    eaBuf[i] = ex;
    atomicAddF(nodeSum + dn * 4 + h, ex);
}

// ---- pass C: g[dst] += T[src] * ea ----------------------------------------

__global__ void edge_accum_g_kernel(const float* __restrict__ T, const float* __restrict__ eaBuf,
                                    const float* __restrict__ nodeSum,
                                    const int* __restrict__ src, const int* __restrict__ dst,
                                    float* __restrict__ g, int total) {
    int i = blockIdx.x * blockDim.x + threadIdx.x;
    if (i >= total) return;
    const int e = i >> 6, j = i & 63, h = j >> 4;
    const int s = src[e], dn = dst[e];
    float ea = eaBuf[e * 4 + h] / nodeSum[dn * 4 + h];
    atomicAddF(g + dn * 64 + j, T[s * 64 + j] * ea);
}

// ---- pass D: per-(node,head) MLP -> fi[N][H][M], Wi staged in LDS ---------

__global__ __launch_bounds__(256)
void node_mlp_kernel(const float* __restrict__ g, const float* __restrict__ t,
                     const float* __restrict__ Wi, const float* __restrict__ bi,
                     const float* __restrict__ weight_i, const float* __restrict__ scale_i,
                     float* __restrict__ fi, int total) {
    __shared__ float sWi[128 * 17];
    __shared__ float sBi[128];
    __shared__ float sWm[128];
    __shared__ float sSc[8];
    for (int i = threadIdx.x; i < 128 * 17; i += 256) sWi[i] = Wi[i];
    for (int i = threadIdx.x; i < 128; i += 256) { sBi[i] = bi[i]; sWm[i] = weight_i[i]; }
    if (threadIdx.x < 8) sSc[threadIdx.x] = __expf(scale_i[threadIdx.x]);
    __syncthreads();

    int nh = blockIdx.x * 256 + threadIdx.x;
    if (nh >= total) return;

    float in[17];
    const float* gp = g + nh * 16;     // (n*4+h)*16 == n*64 + h*16
    #pragma unroll
    for (int j = 0; j < 16; ++j) in[j] = gp[j];
    in[16] = t[nh >> 2];

    #pragma unroll
    for (int m = 0; m < 8; ++m) {
        float z = 0.f;
        for (int dl = 0; dl < 16; ++dl) {
            const int rowi = m * 16 + dl;
            const float* wr = sWi + rowi * 17;
            float a = sBi[rowi];
            #pragma unroll
            for (int k = 0; k < 17; ++k) a += wr[k] * in[k];
            float u = 1.f / (1.f + __expf(-a));            // sigmoid
            z += u * sWm[rowi];
        }
        float sc = sSc[m];
        z /= sc;
        fi[nh * 8 + m] = sc * log1pf(__expf(z));           // scale * softplus
    }
}

// ---- pass E: out[dst] += V[src] * (fi[src].fm[dst]) * ea ------------------

__global__ void edge_out_kernel(const float* __restrict__ V, const float* __restrict__ fi,
                                const float* __restrict__ mArr, const float* __restrict__ eaBuf,
                                const float* __restrict__ nodeSum,
                                const int* __restrict__ src, const int* __restrict__ dst,
                                float* __restrict__ out, int total) {
    int i = blockIdx.x * blockDim.x + threadIdx.x;
    if (i >= total) return;
    const int e = i >> 6, j = i & 63, h = j >> 4;
    const int s = src[e], dn = dst[e];
    float ea = eaBuf[e * 4 + h] / nodeSum[dn * 4 + h];
    const float* fp = fi + (s * 4 + h) * 8;
    const float* fm = mArr + dn * 8;
    float ee = 0.f;
    #pragma unroll
    for (int m = 0; m < 8; ++m) ee += fp[m] * fm[m];
    ee *= ea;
    atomicAddF(out + dn * 64 + j, V[s * 64 + j] * ee);
}

// ---------------------------------------------------------------------------

extern "C" void kernel_launch(void* const* d_in, const int* in_sizes, int n_in,
                              void* d_out, int out_size, void* d_ws, size_t ws_size,
                              hipStream_t stream) {
    const float* x_src = (const float*)d_in[0];
    const float* x_dst = (const float*)d_in[1];
    const float* t     = (const float*)d_in[2];
    const float* mArr  = (const float*)d_in[3];
    const float* ef    = (const float*)d_in[4];
    const float* Wq    = (const float*)d_in[5];
    const float* Wk    = (const float*)d_in[6];
    const float* Wv    = (const float*)d_in[7];
    const float* Wt    = (const float*)d_in[8];
    const float* We    = (const float*)d_in[9];
    const float* Wi    = (const float*)d_in[10];
    const float* bi    = (const float*)d_in[11];
    const float* wgt_i = (const float*)d_in[12];
    const float* scl_i = (const float*)d_in[13];
    const int*   src   = (const int*)d_in[14];
    const int*   dst   = (const int*)d_in[15];
    float* out = (float*)d_out;

    const int N = in_sizes[2];       // t has N elements
    const int E = in_sizes[14];      // src has E elements
    const int NN = N * 64, N4 = N * 4;

    // workspace layout (floats): Q | K | V | T | g | qe | nodeMax | nodeSum | fi | eaBuf
    float* ws      = (float*)d_ws;
    float* Q       = ws;
    float* K       = ws + (size_t)NN;
    float* V       = ws + (size_t)2 * NN;
    float* T       = ws + (size_t)3 * NN;
    float* g       = ws + (size_t)4 * NN;
    float* qe      = ws + (size_t)5 * NN;
    unsigned* nMax = (unsigned*)(ws + (size_t)5 * NN + N4);
    float* nSum    = ws + (size_t)5 * NN + (size_t)2 * N4;
    float* fi      = ws + (size_t)5 * NN + (size_t)3 * N4;
    float* eaBuf   = ws + (size_t)5 * NN + (size_t)3 * N4 + (size_t)N * 32;

    init_kernel<<<(NN + 255) / 256, 256, 0, stream>>>(x_dst, out, g, nMax, nSum, NN, N4);

    dim3 pg((N + 15) / 16, 16);      // 16 = 4 matrices x 4 column tiles
    proj_wmma_kernel<<<pg, 32, 0, stream>>>(x_src, x_dst, Wq, Wk, Wv, Wt, Q, K, V, T, N);

    qe_kernel<<<(N4 + 255) / 256, 256, 0, stream>>>(Q, We, qe, N4);

    const int EH = E * 4;
    edge_logits_kernel<<<(EH + 255) / 256, 256, 0, stream>>>(K, Q, qe, ef, src, dst, eaBuf, nMax, EH);
    edge_exp_kernel<<<(EH + 255) / 256, 256, 0, stream>>>(dst, nMax, eaBuf, nSum, EH);

    const int E64 = E * 64;
    edge_accum_g_kernel<<<(E64 + 255) / 256, 256, 0, stream>>>(T, eaBuf, nSum, src, dst, g, E64);

    node_mlp_kernel<<<(N4 + 255) / 256, 256, 0, stream>>>(g, t, Wi, bi, wgt_i, scl_i, fi, N4);

    edge_out_kernel<<<(E64 + 255) / 256, 256, 0, stream>>>(V, fi, mArr, eaBuf, nSum, src, dst, out, E64);
}